// SheafConvLayer_63067299775041
// MI455X (gfx1250) — compile-verified
//
#include <hip/hip_runtime.h>
#include <hip/hip_bf16.h>

// ---------------- CDNA5 (gfx1250) sheaf-convolution pipeline ----------------
// fp32 WMMA (V_WMMA_F32_16X16X4_F32) for y = emb @ lin_W, TDM tensor_load for
// the lin_W stage, factorized restriction maps (no reverse-edge search),
// L2-resident f32 atomics for the scatter.

typedef float v2f  __attribute__((ext_vector_type(2)));
typedef float v8f  __attribute__((ext_vector_type(8)));
typedef unsigned int u32x4 __attribute__((ext_vector_type(4)));
typedef int i32x4 __attribute__((ext_vector_type(4)));
typedef int i32x8 __attribute__((ext_vector_type(8)));

#define LATENT 64

// ---- K0: zero the diagonal accumulator (must happen every launch) ----
__global__ void k0_zero(float* __restrict__ diag, int n) {
    int i = blockIdx.x * blockDim.x + threadIdx.x;
    if (i < n) diag[i] = 0.0f;
}

// ---- K1: fused GEMM (WMMA f32) + per-node sheaf dot products ----
// Each wave owns a 16-row tile of emb; 4 column tiles of 16, K=64 in 16 steps.
__global__ __launch_bounds__(256) void k1_gemm_ab(
    const float* __restrict__ emb,      // [N,64]
    const float* __restrict__ sheaf_W,  // [128]  (W_left | W_right)
    const float* __restrict__ lin_W,    // [64,64] row-major
    const float* __restrict__ lin_b,    // [64]
    float* __restrict__ y,              // [N,64] out
    float* __restrict__ a_out,          // [N] out
    float* __restrict__ b_out,          // [N] out
    int n_nodes)
{
    __shared__ float ldsRaw[64 * 64];   // 16 KB TDM destination (row-major lin_W)
    __shared__ v2f   ldsB[32 * 80];     // 20 KB K-pair interleaved, pitch-80 padded
    __shared__ float ldsSheaf[128];
    __shared__ float ldsBias[64];

    const int tid = threadIdx.x;

    // ---- TDM: DMA the 64x64 f32 lin_W tile into LDS (wave 0 issues once) ----
    // D# per ISA 8.3/8.4: group0 = {flags, lds_addr, global_addr, type=2},
    // group1 = {data_size=4B, tensor 64x64, tile 64x64, stride0=64, stride1=4096}.
    if (tid < 32) {
        unsigned long long ga = (unsigned long long)lin_W;
        unsigned lraw = (unsigned)(unsigned long long)&ldsRaw[0];
        u32x4 g0;
        g0[0] = 1u;                                    // count=1, user descriptor
        g0[1] = lraw;                                  // lds_addr (bytes)
        g0[2] = (unsigned)(ga & 0xFFFFFFFFull);        // global_addr[31:0]
        g0[3] = (unsigned)((ga >> 32) & 0x1FFFFFFull)  // global_addr[56:32]
              | (2u << 30);                            // type = 2 ("image")
        i32x8 g1;
        g1[0] = (int)(2u << 16);     // workgroup_mask=0 (not in cluster), data_size=4B
        g1[1] = (int)(64u << 16);    // tensor_dim0[15:0] = 64
        g1[2] = (int)(64u << 16);    // tensor_dim0 hi=0 | tensor_dim1[15:0] = 64
        g1[3] = (int)(64u << 16);    // tensor_dim1 hi=0 | tile_dim0 = 64
        g1[4] = 64;                  // tile_dim1 = 64, tile_dim2 = 0
        g1[5] = 64;                  // tensor_dim0_stride = 64 elements
        g1[6] = (int)(4096u << 16);  // stride0 hi=0 | tensor_dim1_stride[15:0] = 4096
        g1[7] = 0;                   // tensor_dim1_stride hi = 0
        i32x4 gz = {0, 0, 0, 0};
#if defined(__clang_major__) && (__clang_major__ >= 23)
        i32x8 gz8 = {0, 0, 0, 0, 0, 0, 0, 0};
        __builtin_amdgcn_tensor_load_to_lds(g0, g1, gz, gz, gz8, 0);
#else
        __builtin_amdgcn_tensor_load_to_lds(g0, g1, gz, gz, 0);
#endif
        __builtin_amdgcn_s_wait_tensorcnt(0);
    }
    if (tid < 128) ldsSheaf[tid] = sheaf_W[tid];
    if (tid < 64)  ldsBias[tid]  = lin_b[tid];
    __syncthreads();

    // Permute raw row-major -> K-pair interleaved so each WMMA B fragment is a
    // single aligned ds_load_b64; pitch 80 float2 keeps the two wave halves in
    // disjoint bank ranges (2*80 = 160 dwords == 32 mod 64).
    for (int t = tid; t < 32 * 64; t += 256) {
        int p = t >> 6, c = t & 63;
        v2f v;
        v.x = ldsRaw[(2 * p)     * 64 + c];
        v.y = ldsRaw[(2 * p + 1) * 64 + c];
        ldsB[p * 80 + c] = v;
    }
    __syncthreads();

    const int wave = tid >> 5;
    const int lane = tid & 31;
    const int mrow = lane & 15;   // M index within tile
    const int h    = lane >> 4;   // K-half selector
    const int tile_row = (blockIdx.x * 8 + wave) * 16;

    // A fragments (ISA 7.12.2, 32-bit A 16x4): element (m,k) lives in
    // lane = m + 16*(k>>1), vgpr = k&1. Per lane: .x = emb[m][4s+2h], .y = +1.
    const int arow   = tile_row + mrow;
    const int arow_c = arow < n_nodes ? arow : n_nodes - 1;  // clamp, keep EXEC uniform
    const float* aptr = emb + (size_t)arow_c * LATENT + 2 * h;
    v2f afrag[16];
    #pragma unroll
    for (int s = 0; s < 16; ++s)
        afrag[s] = *(const v2f*)(aptr + 4 * s);

    v8f acc[4] = {};
    #pragma unroll
    for (int s = 0; s < 16; ++s) {
        #pragma unroll
        for (int j = 0; j < 4; ++j) {
            // B (4x16 f32): lanes 0-15 hold rows K=4s+2h, 4s+2h+1 (mirror of A)
            v2f bfrag = ldsB[(2 * s + h) * 80 + 16 * j + mrow];
            acc[j] = __builtin_amdgcn_wmma_f32_16x16x4_f32(
                false, afrag[s], false, bfrag, (short)0, acc[j], false, false);
        }
    }

    // C/D layout: vgpr i, lane -> row = i + 8*h, col = 16j + mrow
    #pragma unroll
    for (int j = 0; j < 4; ++j) {
        #pragma unroll
        for (int i = 0; i < 8; ++i) {
            int r = tile_row + i + 8 * h;
            int c = 16 * j + mrow;
            if (r < n_nodes) y[(size_t)r * LATENT + c] = acc[j][i] + ldsBias[c];
        }
    }

    // a[i] = emb[i] . W_left, b[i] = emb[i] . W_right, reusing A fragments.
    float pa = 0.0f, pb = 0.0f;
    #pragma unroll
    for (int s = 0; s < 16; ++s) {
        float wl0 = ldsSheaf[4 * s + 2 * h];
        float wl1 = ldsSheaf[4 * s + 2 * h + 1];
        float wr0 = ldsSheaf[64 + 4 * s + 2 * h];
        float wr1 = ldsSheaf[64 + 4 * s + 2 * h + 1];
        pa += afrag[s].x * wl0 + afrag[s].y * wl1;
        pb += afrag[s].x * wr0 + afrag[s].y * wr1;
    }
    pa += __shfl_xor(pa, 16, 32);   // combine the two K-halves
    pb += __shfl_xor(pb, 16, 32);
    if (h == 0 && arow < n_nodes) { a_out[arow] = pa; b_out[arow] = pb; }
}

// ---- K2: per-edge map + diagonal segment sum (atomics into L2) ----
__global__ void k2_edge_maps(const long long* __restrict__ erow,
                             const long long* __restrict__ ecol,
                             const float* __restrict__ a,
                             const float* __restrict__ b,
                             float* __restrict__ m_out,
                             float* __restrict__ diag,
                             int e_cnt)
{
    int e = blockIdx.x * blockDim.x + threadIdx.x;
    if (e >= e_cnt) return;
    int r = (int)erow[e];
    int c = (int)ecol[e];
    float m = tanhf(a[r] + b[c]);
    m_out[e] = m;
    atomicAdd(diag + r, m * m);
}

// ---- K3a: dsi = (diag+1)^-1/2 ----
__global__ void k3_dsi(const float* __restrict__ diag, float* __restrict__ dsi, int n) {
    int i = blockIdx.x * blockDim.x + threadIdx.x;
    if (i < n) dsi[i] = rsqrtf(diag[i] + 1.0f);
}

// ---- K3b: out = emb - (d/(d+1)) * y   (diagonal part of L), float4 ----
__global__ void k3_out_init(const float* __restrict__ emb,
                            const float* __restrict__ y,
                            const float* __restrict__ diag,
                            float* __restrict__ out, int n_nodes)
{
    int i = blockIdx.x * blockDim.x + threadIdx.x;   // over n_nodes*16 float4s
    if (i >= n_nodes * 16) return;
    int node = i >> 4;
    float dm = diag[node];
    float scale = dm / (dm + 1.0f);
    float4 e4 = ((const float4*)emb)[i];
    float4 y4 = ((const float4*)y)[i];
    float4 o;
    o.x = e4.x - scale * y4.x;
    o.y = e4.y - scale * y4.y;
    o.z = e4.z - scale * y4.z;
    o.w = e4.w - scale * y4.w;
    ((float4*)out)[i] = o;
}

// ---- K4: wave-per-edge scatter: out[row] += w_e * y[col] ----
// norm = dsi_r * (-m * m_rev) * dsi_c ; out -= norm*y[col]  =>  += w*y[col]
// with w = dsi_r * m * m_rev * dsi_c.  m_rev = tanh(a[col]+b[row]) (no reverse
// index search needed).  Each lane covers 2 columns (float2) -> contiguous
// atomic addresses across the wave.
__global__ __launch_bounds__(256) void k4_edge_scatter(
    const long long* __restrict__ erow, const long long* __restrict__ ecol,
    const float* __restrict__ a, const float* __restrict__ b,
    const float* __restrict__ m_arr, const float* __restrict__ dsi,
    const float* __restrict__ y, float* __restrict__ out, int e_cnt)
{
    int e    = blockIdx.x * 8 + (threadIdx.x >> 5);
    int lane = threadIdx.x & 31;
    if (e >= e_cnt) return;
    int r = (int)erow[e];
    int c = (int)ecol[e];
    float m    = m_arr[e];
    float mrev = tanhf(a[c] + b[r]);
    float w    = dsi[r] * m * mrev * dsi[c];
    const float2 y2 = *(const float2*)(y + (size_t)c * LATENT + 2 * lane);
    float* op = out + (size_t)r * LATENT + 2 * lane;
    atomicAdd(op,     w * y2.x);
    atomicAdd(op + 1, w * y2.y);
}

extern "C" void kernel_launch(void* const* d_in, const int* in_sizes, int n_in,
                              void* d_out, int out_size, void* d_ws, size_t ws_size,
                              hipStream_t stream) {
    const float*     emb     = (const float*)d_in[0];      // [N,64]
    const long long* eidx    = (const long long*)d_in[1];  // [2,E] int64
    const float*     sheaf_W = (const float*)d_in[2];      // [128]
    const float*     lin_W   = (const float*)d_in[3];      // [64,64]
    const float*     lin_b   = (const float*)d_in[4];      // [64]
    float*           out     = (float*)d_out;              // [N,64]

    const int n_nodes = in_sizes[0] / LATENT;
    const int e_cnt   = in_sizes[1] / 2;
    const long long* erow = eidx;
    const long long* ecol = eidx + e_cnt;

    // workspace layout (floats)
    float* ws   = (float*)d_ws;
    float* y    = ws;                               // N*64
    float* a    = y    + (size_t)n_nodes * LATENT;  // N
    float* b    = a    + n_nodes;                   // N
    float* diag = b    + n_nodes;                   // N
    float* dsi  = diag + n_nodes;                   // N
    float* marr = dsi  + n_nodes;                   // E

    k0_zero<<<(n_nodes + 255) / 256, 256, 0, stream>>>(diag, n_nodes);

    k1_gemm_ab<<<(n_nodes + 127) / 128, 256, 0, stream>>>(
        emb, sheaf_W, lin_W, lin_b, y, a, b, n_nodes);

    k2_edge_maps<<<(e_cnt + 255) / 256, 256, 0, stream>>>(
        erow, ecol, a, b, marr, diag, e_cnt);

    k3_dsi<<<(n_nodes + 255) / 256, 256, 0, stream>>>(diag, dsi, n_nodes);

    k3_out_init<<<(n_nodes * 16 + 255) / 256, 256, 0, stream>>>(
        emb, y, diag, out, n_nodes);

    k4_edge_scatter<<<(e_cnt + 7) / 8, 256, 0, stream>>>(
        erow, ecol, a, b, marr, dsi, y, out, e_cnt);
}